// MV_MOE_82952998355169
// MI455X (gfx1250) — compile-verified
//
#include <hip/hip_runtime.h>
#include <hip/hip_bf16.h>

typedef float v2f __attribute__((ext_vector_type(2)));
typedef float v8f __attribute__((ext_vector_type(8)));

#define S_TOK   8192
#define FD      512
#define E_NUM   16
#define CD      256

#define ACT_NONE  0
#define ACT_RELU  1
#define ACT_LRELU 2

#define WMMA_F32(a, b, c) \
    __builtin_amdgcn_wmma_f32_16x16x4_f32(false, (a), false, (b), (short)0, (c), false, false)

struct Frags { v2f a0, a1, a2, a3, b0, b1; };

static __device__ __forceinline__ Frags mvmoe_load_frags(
    const float* __restrict__ A, unsigned aoff, unsigned aS16,
    const float* __restrict__ Bp, unsigned boff)
{
    Frags f;
    f.a0 = *(const v2f*)(A + aoff);
    f.a1 = *(const v2f*)(A + aoff + aS16);
    f.a2 = *(const v2f*)(A + aoff + 2 * aS16);
    f.a3 = *(const v2f*)(A + aoff + 3 * aS16);
    f.b0 = *(const v2f*)(Bp + boff);
    f.b1 = *(const v2f*)(Bp + boff + 32);
    return f;
}

static __device__ __forceinline__ void mvmoe_mma8(
    const Frags& f, v8f& c00, v8f& c01, v8f& c10, v8f& c11,
    v8f& c20, v8f& c21, v8f& c30, v8f& c31)
{
    c00 = WMMA_F32(f.a0, f.b0, c00); c01 = WMMA_F32(f.a0, f.b1, c01);
    c10 = WMMA_F32(f.a1, f.b0, c10); c11 = WMMA_F32(f.a1, f.b1, c11);
    c20 = WMMA_F32(f.a2, f.b0, c20); c21 = WMMA_F32(f.a2, f.b1, c21);
    c30 = WMMA_F32(f.a3, f.b0, c30); c31 = WMMA_F32(f.a3, f.b1, c31);
}

// ---------------------------------------------------------------------------
// fp32 WMMA GEMM over pair-packed B: C = act(A[M,K] @ B + bias); B stored as
// P[k/2][N][2] so each B fragment is one b64 load. Per wave: 64(M) x 32(N)
// tile -> 8 WMMAs per K-step from 6 b64 loads. Double-buffered (ping-pong)
// pipeline, manually unrolled x2 so there are no register-rotation moves.
// Requires K%8==0, N%32==0, M%64==0; act is a template param (branch-free
// epilogue); all offsets 32-bit from uniform bases (saddr+voffset loads).
// ---------------------------------------------------------------------------
template <int ACT>
__global__ __launch_bounds__(256) void mvmoe_wmma_gemm(
    const float* __restrict__ A, int lda,
    const float* __restrict__ Bp,      // packed [K/2][N][2]
    const float* __restrict__ bias,    // [N]
    float* __restrict__ C,             // [M][N]
    int M, int K, int N)
{
    const int tiles_n = N >> 5;
    const int wave = threadIdx.x >> 5;
    const int lane = threadIdx.x & 31;
    const int tiles_m = M >> 6;
    const int tileId = blockIdx.x * 8 + wave;
    if (tileId >= tiles_m * tiles_n) return;      // whole-wave uniform exit
    const int mt = tileId / tiles_n;
    const int nt = tileId - mt * tiles_n;

    const unsigned half = (unsigned)(lane >> 4);
    const unsigned lm   = (unsigned)(lane & 15);
    const unsigned n0   = (unsigned)nt * 32u + lm;      // 2nd column = n0 + 16

    unsigned aoff = ((unsigned)mt * 64u + lm) * (unsigned)lda + 2u * half;
    unsigned boff = (half * (unsigned)N + n0) * 2u;
    const unsigned aS16  = 16u * (unsigned)lda;
    const unsigned bStep = 4u * (unsigned)N;

    v8f c00 = {}, c01 = {}, c10 = {}, c11 = {};
    v8f c20 = {}, c21 = {}, c30 = {}, c31 = {};

    Frags f0 = mvmoe_load_frags(A, aoff, aS16, Bp, boff);   // K-step 0
    const int its = ((K >> 2) - 2) >> 1;                    // K%8==0 -> exact
    for (int i = 0; i < its; ++i) {
        aoff += 4u; boff += bStep;
        Frags f1 = mvmoe_load_frags(A, aoff, aS16, Bp, boff);
        mvmoe_mma8(f0, c00, c01, c10, c11, c20, c21, c30, c31);
        aoff += 4u; boff += bStep;
        f0 = mvmoe_load_frags(A, aoff, aS16, Bp, boff);
        mvmoe_mma8(f1, c00, c01, c10, c11, c20, c21, c30, c31);
    }
    aoff += 4u; boff += bStep;
    Frags fl = mvmoe_load_frags(A, aoff, aS16, Bp, boff);   // last K-step
    mvmoe_mma8(f0, c00, c01, c10, c11, c20, c21, c30, c31);
    mvmoe_mma8(fl, c00, c01, c10, c11, c20, c21, c30, c31);

    const float bn0 = bias[n0];
    const float bn1 = bias[n0 + 16];

    v8f cc[8] = {c00, c01, c10, c11, c20, c21, c30, c31};
    #pragma unroll
    for (int sub = 0; sub < 4; ++sub) {
        #pragma unroll
        for (int col = 0; col < 2; ++col) {
            const unsigned base =
                ((unsigned)mt * 64u + (unsigned)sub * 16u + 8u * half) * (unsigned)N
                + n0 + (unsigned)col * 16u;
            const float bb = col ? bn1 : bn0;
            #pragma unroll
            for (int j = 0; j < 8; ++j) {
                float v = cc[sub * 2 + col][j] + bb;
                if (ACT == ACT_RELU)       v = fmaxf(v, 0.0f);
                else if (ACT == ACT_LRELU) v = v >= 0.0f ? v : 0.01f * v;
                C[base + (unsigned)j * (unsigned)N] = v;
            }
        }
    }
}

// Zero-pad + pair-pack: out[kp][n][r] = in[2kp+r][n] (0 outside Rin x Cin)
__global__ __launch_bounds__(256) void mvmoe_padpack(
    const float* __restrict__ in, int Rin, int Cin,
    float* __restrict__ out, int Rp, int Cp)
{
    const int g = blockIdx.x * 256 + threadIdx.x;
    if (g >= Rp * Cp) return;
    const int r  = g & 1;
    const int t  = g >> 1;
    const int kp = t / Cp;
    const int n  = t - kp * Cp;
    const int k  = 2 * kp + r;
    out[g] = (k < Rin && n < Cin) ? in[(size_t)k * Cin + n] : 0.0f;
}

// Zero-pad copy (used for biases): out[Rp][Cp] <- in[Rin][Cin]
__global__ __launch_bounds__(256) void mvmoe_pad2d(
    const float* __restrict__ in, int Rin, int Cin,
    float* __restrict__ out, int Rp, int Cp)
{
    const int g = blockIdx.x * 256 + threadIdx.x;
    if (g >= Rp * Cp) return;
    const int r = g / Cp, c = g - r * Cp;
    out[g] = (r < Rin && c < Cin) ? in[(size_t)r * Cin + c] : 0.0f;
}

// ---------------------------------------------------------------------------
// Router: select = h @ Wr + noise, top-1 gate/argmax, deterministic per-block
// partial sums of select (density_proxy) and one-hot counts (density).
// ---------------------------------------------------------------------------
__global__ __launch_bounds__(256) void mvmoe_router(
    const float* __restrict__ hbase, const float* __restrict__ noise1,
    const float* __restrict__ noise2, const float* __restrict__ Wr,
    int* __restrict__ idx, float* __restrict__ gate,
    float* __restrict__ proxyPart, float* __restrict__ countPart)
{
    const int v = blockIdx.y;
    const int s = blockIdx.x * 256 + threadIdx.x;
    const float* h     = hbase + (size_t)v * S_TOK * FD + (size_t)s * FD;
    const float* noise = (v == 0 ? noise1 : noise2) + (size_t)s * E_NUM;

    float acc[E_NUM];
    #pragma unroll
    for (int e = 0; e < E_NUM; ++e) acc[e] = noise[e];
    for (int k = 0; k < FD; ++k) {
        const float hv = h[k];
        const float* wr = Wr + k * E_NUM;
        #pragma unroll
        for (int e = 0; e < E_NUM; ++e) acc[e] += hv * wr[e];
    }

    float best = acc[0]; int bi = 0;
    #pragma unroll
    for (int e = 1; e < E_NUM; ++e)
        if (acc[e] > best) { best = acc[e]; bi = e; }   // first-max == argmax
    idx[v * S_TOK + s]  = bi;
    gate[v * S_TOK + s] = best;

    __shared__ float ssel[256 * E_NUM];
    __shared__ int   sidx[256];
    #pragma unroll
    for (int e = 0; e < E_NUM; ++e) ssel[threadIdx.x * E_NUM + e] = acc[e];
    sidx[threadIdx.x] = bi;
    __syncthreads();
    if (threadIdx.x < E_NUM) {
        const int e = threadIdx.x;
        float ps = 0.0f, cnt = 0.0f;
        for (int t = 0; t < 256; ++t) {          // fixed order -> deterministic
            ps  += ssel[t * E_NUM + e];
            cnt += (sidx[t] == e) ? 1.0f : 0.0f;
        }
        proxyPart[(v * 32 + blockIdx.x) * E_NUM + e] = ps;
        countPart[(v * 32 + blockIdx.x) * E_NUM + e] = cnt;
    }
}

// ein stage 1: 128-token chunks accumulated per-expert in LDS; each thread
// owns one feature column -> race-free and deterministic.
__global__ __launch_bounds__(512) void mvmoe_ein_part(
    const float* __restrict__ hbase, const int* __restrict__ idx,
    const float* __restrict__ gate, float* __restrict__ einp)
{
    const int v = blockIdx.y, chunk = blockIdx.x;
    __shared__ float part[E_NUM * FD];
    for (int i = threadIdx.x; i < E_NUM * FD; i += 512) part[i] = 0.0f;
    __syncthreads();
    const float* h  = hbase + (size_t)v * S_TOK * FD;
    const int*   id = idx  + v * S_TOK;
    const float* g  = gate + v * S_TOK;
    const int t0 = chunk * 128;
    for (int t = 0; t < 128; ++t) {
        const int s = t0 + t;
        const int e = id[s];
        if (g[s] != 0.0f)    // dispatch = (combine != 0)
            part[e * FD + threadIdx.x] += h[(size_t)s * FD + threadIdx.x];
    }
    __syncthreads();
    float* dst = einp + (size_t)(v * 64 + chunk) * E_NUM * FD;
    for (int i = threadIdx.x; i < E_NUM * FD; i += 512) dst[i] = part[i];
}

__global__ __launch_bounds__(512) void mvmoe_ein_reduce(
    const float* __restrict__ einp, float* __restrict__ ein)
{
    const int v = blockIdx.y, e = blockIdx.x;
    float acc = 0.0f;
    for (int c = 0; c < 64; ++c)
        acc += einp[(size_t)(v * 64 + c) * E_NUM * FD + e * FD + threadIdx.x];
    ein[(v * E_NUM + e) * FD + threadIdx.x] = acc;
}

// Batched per-expert MLP layer, both views fused (weights read once):
// out[v][e][j] = act( sum_k in[v][e][k] * W[e][k][j] + b[e][j] )
__global__ __launch_bounds__(256) void mvmoe_expert_layer(
    const float* __restrict__ in, const float* __restrict__ W,
    const float* __restrict__ bvec, float* __restrict__ out,
    int K, int J, int act)
{
    __shared__ float s0[2048], s1[2048];
    const int e = blockIdx.y;
    for (int k = threadIdx.x; k < K; k += 256) {
        s0[k] = in[(size_t)(0 * E_NUM + e) * K + k];
        s1[k] = in[(size_t)(1 * E_NUM + e) * K + k];
    }
    __syncthreads();
    const int j = blockIdx.x * 256 + threadIdx.x;
    if (j >= J) return;
    float a0 = 0.0f, a1 = 0.0f;
    const float* We = W + (size_t)e * K * J + j;
    for (int k = 0; k < K; ++k) {
        const float w = We[(size_t)k * J];
        a0 += s0[k] * w;
        a1 += s1[k] * w;
    }
    const float bb = bvec[(size_t)e * J + j];
    a0 += bb; a1 += bb;
    if (act == ACT_RELU) { a0 = fmaxf(a0, 0.0f); a1 = fmaxf(a1, 0.0f); }
    out[(size_t)(0 * E_NUM + e) * J + j] = a0;
    out[(size_t)(1 * E_NUM + e) * J + j] = a1;
}

// fused[s, c] = gate[v][s] * eo[v][idx[v][s]][c%256], v = c/256
__global__ __launch_bounds__(256) void mvmoe_scatter(
    const float* __restrict__ eo, const int* __restrict__ idx,
    const float* __restrict__ gate, float* __restrict__ fused)
{
    const int gidx = blockIdx.x * 256 + threadIdx.x;
    const int s = gidx >> 9;
    const int c = gidx & 511;
    const int v = c >> 8;
    const int cc = c & 255;
    const int e = idx[v * S_TOK + s];
    fused[gidx] = gate[v * S_TOK + s] * eo[(size_t)((v * E_NUM + e) << 8) + cc];
}

__global__ void mvmoe_balance(
    const float* __restrict__ proxyPart, const float* __restrict__ countPart,
    float* __restrict__ out_bal, float* __restrict__ out_dist)
{
    if (threadIdx.x != 0 || blockIdx.x != 0) return;
    float total = 0.0f;
    for (int v = 0; v < 2; ++v) {
        float bl = 0.0f;
        for (int e = 0; e < E_NUM; ++e) {
            float ps = 0.0f, cs = 0.0f;
            for (int b = 0; b < 32; ++b) {
                ps += proxyPart[(v * 32 + b) * E_NUM + e];
                cs += countPart[(v * 32 + b) * E_NUM + e];
            }
            bl += (ps / (float)S_TOK) * (cs / (float)S_TOK);
        }
        total += bl * (float)E_NUM;   // (sum/E) * E^2 = sum * E
    }
    *out_bal  = total;
    *out_dist = 0.0f;
}

// ---------------------------------------------------------------------------
extern "C" void kernel_launch(void* const* d_in, const int* in_sizes, int n_in,
                              void* d_out, int out_size, void* d_ws, size_t ws_size,
                              hipStream_t stream) {
    const float* x1  = (const float*)d_in[0];
    const float* x2  = (const float*)d_in[1];
    const float* nz1 = (const float*)d_in[2];
    const float* nz2 = (const float*)d_in[3];
    const float* Wp1 = (const float*)d_in[4];  const float* bp1 = (const float*)d_in[5];
    const float* Wp2 = (const float*)d_in[6];  const float* bp2 = (const float*)d_in[7];
    const float* Wr  = (const float*)d_in[8];
    const float* We1 = (const float*)d_in[9];   const float* be1 = (const float*)d_in[10];
    const float* We2 = (const float*)d_in[11];  const float* be2 = (const float*)d_in[12];
    const float* We3 = (const float*)d_in[13];  const float* be3 = (const float*)d_in[14];
    const float* We4 = (const float*)d_in[15];  const float* be4 = (const float*)d_in[16];
    const float* Wd1[4] = {(const float*)d_in[17], (const float*)d_in[19],
                           (const float*)d_in[21], (const float*)d_in[23]};
    const float* bd1[4] = {(const float*)d_in[18], (const float*)d_in[20],
                           (const float*)d_in[22], (const float*)d_in[24]};
    const float* Wd2[4] = {(const float*)d_in[25], (const float*)d_in[27],
                           (const float*)d_in[29], (const float*)d_in[31]};
    const float* bd2[4] = {(const float*)d_in[26], (const float*)d_in[28],
                           (const float*)d_in[30], (const float*)d_in[32]};

    float* out = (float*)d_out;
    float* fused = out;                           // [8192][512]
    float* r1    = out + (size_t)S_TOK * 512;     // [8192][2048]
    float* r2    = r1  + (size_t)S_TOK * 2048;    // [8192][1024]
    float* bal   = r2  + (size_t)S_TOK * 1024;
    float* dist  = bal + 1;

    // ---- workspace layout (float offsets, all even -> 8B aligned) ----
    float* ws = (float*)d_ws;
    // persistent through decoder phase: packed+padded weights/biases + t_b
    float* pW1a = ws;                         // packed [512][2016]
    float* pW1b = ws + 1032192;               // packed [512][2016]
    float* pW2a = ws + 2064384;               // packed [2016][512]
    float* pW2b = ws + 3096576;               // packed [2016][512]
    float* pW3a = ws + 4128768;               // packed [512][512]
    float* pW3b = ws + 4390912;               // packed [512][512]
    float* pW4a = ws + 4653056;               // packed [512][2048]
    float* pW4b = ws + 5701632;               // packed [512][1024]
    float* pb1a = ws + 6225920;               // [2016]
    float* pb1b = ws + 6227936;               // [2016]
    float* pb2a = ws + 6229952;               // [512]
    float* pb2b = ws + 6230464;               // [512]
    float* pb3a = ws + 6230976;               // [512]
    float* pb3b = ws + 6231488;               // [512]
    float* t_b  = ws + 6232000;               // [8192][512]
    // MoE-phase buffers, all dead after scatter -> aliased by t_a below
    float* base2  = ws + 10426304;
    float* h      = base2;                    // [2][8192][512]
    int*   idx    = (int*)(base2 + 8388608);  // [2][8192]
    float* gate   = base2 + 8404992;          // [2][8192]
    float* proxyP = base2 + 8421376;          // [2][32][16]
    float* countP = base2 + 8422400;          // [2][32][16]
    float* einp   = base2 + 8423424;          // [2][64][16][512]
    float* ein    = base2 + 9472000;          // [2][16][512]
    float* ea1    = base2 + 9488384;          // [2][16][500]
    float* ea2    = base2 + 9504384;          // [2][16][500]
    float* ea3    = base2 + 9520384;          // [2][16][2000]
    float* eo     = base2 + 9584384;          // [2][16][256]
    float* pWp1   = base2 + 9592576;          // packed [2048][512]
    float* pWp2   = base2 + 10641152;         // packed [1024][512]  (ends 11,165,440)
    float* t_a    = base2;                    // [8192][2016] (decoder phase only)
    float* t_c    = t_a;                      // [8192][512], reuses t_a region

    auto pack = [&](const float* in, int Rin, int Cin, float* o, int Rp, int Cp) {
        mvmoe_padpack<<<dim3((Rp * Cp + 255) / 256), dim3(256), 0, stream>>>(in, Rin, Cin, o, Rp, Cp);
    };
    auto padb = [&](const float* in, int Cin, float* o, int Cp) {
        mvmoe_pad2d<<<dim3((Cp + 255) / 256), dim3(256), 0, stream>>>(in, 1, Cin, o, 1, Cp);
    };
    auto gemm = [&](const float* A, int lda, const float* Bp, const float* bias,
                    float* C, int K, int N, int act) {
        const int total  = (S_TOK / 64) * (N / 32);
        const int blocks = (total + 7) / 8;
        if (act == ACT_NONE)
            mvmoe_wmma_gemm<ACT_NONE><<<dim3(blocks), dim3(256), 0, stream>>>(
                A, lda, Bp, bias, C, S_TOK, K, N);
        else if (act == ACT_RELU)
            mvmoe_wmma_gemm<ACT_RELU><<<dim3(blocks), dim3(256), 0, stream>>>(
                A, lda, Bp, bias, C, S_TOK, K, N);
        else
            mvmoe_wmma_gemm<ACT_LRELU><<<dim3(blocks), dim3(256), 0, stream>>>(
                A, lda, Bp, bias, C, S_TOK, K, N);
    };

    // 0) pack (and zero-pad where ragged) all GEMM weights; pad ragged biases
    pack(Wp1,   2048,  512, pWp1, 2048,  512);
    pack(Wp2,   1024,  512, pWp2, 1024,  512);
    pack(Wd1[0], 512, 2000, pW1a,  512, 2016);
    pack(Wd2[0], 512, 2000, pW1b,  512, 2016);
    pack(Wd1[1],2000,  500, pW2a, 2016,  512);
    pack(Wd2[1],2000,  500, pW2b, 2016,  512);
    pack(Wd1[2], 500,  500, pW3a,  512,  512);
    pack(Wd2[2], 500,  500, pW3b,  512,  512);
    pack(Wd1[3], 500, 2048, pW4a,  512, 2048);
    pack(Wd2[3], 500, 1024, pW4b,  512, 1024);
    padb(bd1[0], 2000, pb1a, 2016);
    padb(bd2[0], 2000, pb1b, 2016);
    padb(bd1[1],  500, pb2a,  512);
    padb(bd2[1],  500, pb2b,  512);
    padb(bd1[2],  500, pb3a,  512);
    padb(bd2[2],  500, pb3b,  512);

    // 1) pre-projections
    gemm(x1, 2048, pWp1, bp1, h,                       2048, 512, ACT_NONE);
    gemm(x2, 1024, pWp2, bp2, h + (size_t)S_TOK * 512, 1024, 512, ACT_NONE);

    // 2) router (both views)
    mvmoe_router<<<dim3(32, 2), dim3(256), 0, stream>>>(h, nz1, nz2, Wr, idx, gate, proxyP, countP);

    // 3) expert input = segmented sum of assigned tokens (deterministic)
    mvmoe_ein_part<<<dim3(64, 2), dim3(512), 0, stream>>>(h, idx, gate, einp);
    mvmoe_ein_reduce<<<dim3(16, 2), dim3(512), 0, stream>>>(einp, ein);

    // 4) expert MLP, both views fused (weights read once)
    mvmoe_expert_layer<<<dim3(2, 16), dim3(256), 0, stream>>>(ein, We1, be1, ea1,  512,  500, ACT_RELU);
    mvmoe_expert_layer<<<dim3(2, 16), dim3(256), 0, stream>>>(ea1, We2, be2, ea2,  500,  500, ACT_RELU);
    mvmoe_expert_layer<<<dim3(8, 16), dim3(256), 0, stream>>>(ea2, We3, be3, ea3,  500, 2000, ACT_RELU);
    mvmoe_expert_layer<<<dim3(1, 16), dim3(256), 0, stream>>>(ea3, We4, be4, eo,  2000,  256, ACT_NONE);

    // 5) gather expert outputs -> fused (first model output)
    mvmoe_scatter<<<dim3((S_TOK * 512) / 256), dim3(256), 0, stream>>>(eo, idx, gate, fused);

    // 6) balance + dist outputs (before MoE-phase buffers get reused)
    mvmoe_balance<<<dim3(1), dim3(64), 0, stream>>>(proxyP, countP, bal, dist);

    // 7) decoder 1: 512 -> 2016(2000) -> 512(500) -> 512(500) -> 2048
    gemm(fused, 512, pW1a, pb1a,   t_a,  512, 2016, ACT_LRELU);
    gemm(t_a,  2016, pW2a, pb2a,   t_b, 2016,  512, ACT_LRELU);
    gemm(t_b,   512, pW3a, pb3a,   t_c,  512,  512, ACT_LRELU);
    gemm(t_c,   512, pW4a, bd1[3], r1,   512, 2048, ACT_NONE);

    // 8) decoder 2: 512 -> 2016(2000) -> 512(500) -> 512(500) -> 1024
    gemm(fused, 512, pW1b, pb1b,   t_a,  512, 2016, ACT_LRELU);
    gemm(t_a,  2016, pW2b, pb2b,   t_b, 2016,  512, ACT_LRELU);
    gemm(t_b,   512, pW3b, pb3b,   t_c,  512,  512, ACT_LRELU);
    gemm(t_c,   512, pW4b, bd2[3], r2,   512, 1024, ACT_NONE);

    (void)in_sizes; (void)n_in; (void)out_size; (void)ws_size;
}